// HAN_48524540510789
// MI455X (gfx1250) — compile-verified
//
#include <hip/hip_runtime.h>
#include <hip/hip_bf16.h>
#include <math.h>

#define N_NODES   100000
#define N_EDGES   1000000
#define N_PATHS   3
#define IN_DIM    256
#define HF        64        // HEADS*OUT_DIM
#define HEADS     8
#define OUT_DIM   8
#define SEM_DIM   128
#define N_CLASSES 16
#define NEG_SLOPE 0.2f
#define MTOT      (N_EDGES + N_NODES)   // messages incl. self loops

typedef __attribute__((ext_vector_type(2))) float v2f;
typedef __attribute__((ext_vector_type(8))) float v8f;

// ---------------------------------------------------------------------------
// Kernel 1: h = x @ W   (100000x256 @ 256x64) via V_WMMA_F32_16X16X4_F32
// Block = 128 threads = 4 waves; each wave computes one 16x16 tile of the
// 16x64 row-block. N_NODES % 16 == 0 so no row remainder.
// ---------------------------------------------------------------------------
__global__ void gemm_h_kernel(const float* __restrict__ x,
                              const float* __restrict__ W,
                              float* __restrict__ h) {
    const int m0   = blockIdx.x * 16;
    const int wave = threadIdx.x >> 5;
    const int lane = threadIdx.x & 31;
    const int n0   = wave * 16;
    const int half = lane >> 4;
    const int l16  = lane & 15;
    const int arow = m0 + l16;
    const int bcol = n0 + l16;

    v8f c = {};
    #pragma unroll 4
    for (int k0 = 0; k0 < IN_DIM; k0 += 4) {
        const int ka = k0 + half * 2;
        v2f a, b;
        a.x = x[(size_t)arow * IN_DIM + ka];
        a.y = x[(size_t)arow * IN_DIM + ka + 1];
        b.x = W[(size_t)ka * HF + bcol];
        b.y = W[(size_t)(ka + 1) * HF + bcol];
        c = __builtin_amdgcn_wmma_f32_16x16x4_f32(
                false, a, false, b, (short)0, c, false, false);
    }
    #pragma unroll
    for (int j = 0; j < 8; ++j) {
        const int row = m0 + j + half * 8;
        h[(size_t)row * HF + bcol] = c[j];
    }
}

// ---------------------------------------------------------------------------
// Kernel 2: per-node attention coefficients a_s[n,h], a_d[n,h]
// ---------------------------------------------------------------------------
__global__ void att_coeff_kernel(const float* __restrict__ h,
                                 const float* __restrict__ a_src,
                                 const float* __restrict__ a_dst,
                                 float* __restrict__ as_out,
                                 float* __restrict__ ad_out) {
    const int n = blockIdx.x * blockDim.x + threadIdx.x;
    if (n >= N_NODES) return;
    #pragma unroll
    for (int hh = 0; hh < HEADS; ++hh) {
        float s = 0.f, d = 0.f;
        #pragma unroll
        for (int f = 0; f < OUT_DIM; ++f) {
            const float v = h[(size_t)n * HF + hh * OUT_DIM + f];
            s += v * a_src[hh * OUT_DIM + f];
            d += v * a_dst[hh * OUT_DIM + f];
        }
        as_out[n * HEADS + hh] = s;
        ad_out[n * HEADS + hh] = d;
    }
}

// ---------------------------------------------------------------------------
// Kernel 3: init accumulators (maxv -> INT_MIN encoded, sums/embeds -> 0)
// ---------------------------------------------------------------------------
__global__ void init_kernel(int* __restrict__ maxv, float* __restrict__ sumv,
                            float* __restrict__ embeds, float* __restrict__ scores) {
    const size_t idx = (size_t)blockIdx.x * blockDim.x + threadIdx.x;
    if (idx < (size_t)N_PATHS * N_NODES * HF) embeds[idx] = 0.f;
    if (idx < (size_t)N_PATHS * N_NODES * HEADS) {
        maxv[idx] = (int)0x80000000;
        sumv[idx] = 0.f;
    }
    if (idx < N_PATHS) scores[idx] = 0.f;
}

__device__ __forceinline__ int enc_f32(float f) {
    int i = __float_as_int(f);
    return (i >= 0) ? i : (i ^ 0x7fffffff);
}
__device__ __forceinline__ float dec_f32(int i) {
    return __int_as_float((i >= 0) ? i : (i ^ 0x7fffffff));
}
__device__ __forceinline__ float leaky(float v) {
    return v > 0.f ? v : NEG_SLOPE * v;
}

// ---------------------------------------------------------------------------
// Kernel 4: segment-max over incoming messages (one thread per (msg,head))
// ---------------------------------------------------------------------------
__global__ void edge_max_kernel(const int* __restrict__ src, const int* __restrict__ dst,
                                const float* __restrict__ a_s, const float* __restrict__ a_d,
                                int* __restrict__ maxv) {
    const size_t idx = (size_t)blockIdx.x * blockDim.x + threadIdx.x;
    if (idx >= (size_t)MTOT * HEADS) return;
    const int e  = (int)(idx >> 3);
    const int hh = (int)(idx & 7);
    int s, d;
    if (e < N_EDGES) { s = src[e]; d = dst[e]; } else { s = d = e - N_EDGES; }
    const float v = leaky(a_s[s * HEADS + hh] + a_d[d * HEADS + hh]);
    atomicMax(&maxv[d * HEADS + hh], enc_f32(v));
}

// ---------------------------------------------------------------------------
// Kernel 5: segment-sum of exp(e - max); optionally stash w sequentially so
// the accumulate pass replaces random node gathers with a streaming read.
// ---------------------------------------------------------------------------
__global__ void edge_sum_kernel(const int* __restrict__ src, const int* __restrict__ dst,
                                const float* __restrict__ a_s, const float* __restrict__ a_d,
                                const int* __restrict__ maxv, float* __restrict__ sumv,
                                float* __restrict__ wbuf /* may be null */) {
    const size_t idx = (size_t)blockIdx.x * blockDim.x + threadIdx.x;
    if (idx >= (size_t)MTOT * HEADS) return;
    const int e  = (int)(idx >> 3);
    const int hh = (int)(idx & 7);
    int s, d;
    if (e < N_EDGES) { s = src[e]; d = dst[e]; } else { s = d = e - N_EDGES; }
    const float v = leaky(a_s[s * HEADS + hh] + a_d[d * HEADS + hh]);
    const float m = dec_f32(maxv[d * HEADS + hh]);
    const float w = __expf(v - m);
    atomicAdd(&sumv[d * HEADS + hh], w);
    if (wbuf) wbuf[idx] = w;
}

// ---------------------------------------------------------------------------
// Kernel 6a (fast path): embeds[dst] += w * h[src], w streamed from wbuf.
// One thread per (msg, quad-of-features): float4 load of h, 4 atomics.
// ---------------------------------------------------------------------------
__global__ void edge_acc_w_kernel(const int* __restrict__ src, const int* __restrict__ dst,
                                  const float* __restrict__ wbuf,
                                  const float* __restrict__ h, float* __restrict__ embeds) {
    const size_t idx = (size_t)blockIdx.x * blockDim.x + threadIdx.x;
    if (idx >= (size_t)MTOT * (HF / 4)) return;
    const int e  = (int)(idx >> 4);
    const int q  = (int)(idx & 15);       // quad index: features q*4..q*4+3
    const int f0 = q * 4;
    const int hh = q >> 1;                // head = f0/8
    int s, d;
    if (e < N_EDGES) { s = src[e]; d = dst[e]; } else { s = d = e - N_EDGES; }
    const float w = wbuf[(size_t)e * HEADS + hh];
    const float4 hv = *(const float4*)&h[(size_t)s * HF + f0];
    float* eb = &embeds[(size_t)d * HF + f0];
    atomicAdd(eb + 0, w * hv.x);
    atomicAdd(eb + 1, w * hv.y);
    atomicAdd(eb + 2, w * hv.z);
    atomicAdd(eb + 3, w * hv.w);
}

// ---------------------------------------------------------------------------
// Kernel 6b (fallback, recomputes w): one thread per (msg, feature)
// ---------------------------------------------------------------------------
__global__ void edge_acc_kernel(const int* __restrict__ src, const int* __restrict__ dst,
                                const float* __restrict__ a_s, const float* __restrict__ a_d,
                                const int* __restrict__ maxv,
                                const float* __restrict__ h, float* __restrict__ embeds) {
    const size_t idx = (size_t)blockIdx.x * blockDim.x + threadIdx.x;
    if (idx >= (size_t)MTOT * HF) return;
    const int e  = (int)(idx >> 6);
    const int f  = (int)(idx & 63);
    const int hh = f >> 3;
    int s, d;
    if (e < N_EDGES) { s = src[e]; d = dst[e]; } else { s = d = e - N_EDGES; }
    const float v = leaky(a_s[s * HEADS + hh] + a_d[d * HEADS + hh]);
    const float m = dec_f32(maxv[d * HEADS + hh]);
    const float w = __expf(v - m);
    atomicAdd(&embeds[(size_t)d * HF + f], w * h[(size_t)s * HF + f]);
}

// ---------------------------------------------------------------------------
// Kernel 7: normalize by denom and add bias
// ---------------------------------------------------------------------------
__global__ void edge_fin_kernel(const float* __restrict__ sumv,
                                const float* __restrict__ bias,
                                float* __restrict__ embeds) {
    const size_t idx = (size_t)blockIdx.x * blockDim.x + threadIdx.x;
    if (idx >= (size_t)N_NODES * HF) return;
    const int n = (int)(idx >> 6);
    const int f = (int)(idx & 63);
    const float denom = sumv[n * HEADS + (f >> 3)] + 1e-16f;
    embeds[idx] = embeds[idx] / denom + bias[f];
}

// ---------------------------------------------------------------------------
// Kernel 8: semantic scores  scores[p] += sum_n tanh(embeds_p[n]@semW + b) . q
// WMMA f32 16x16x4: block = 8 waves, wave w handles a 16-col tile of SEM_DIM.
// ---------------------------------------------------------------------------
__global__ void scores_kernel(const float* __restrict__ embeds_p,
                              const float* __restrict__ semW,
                              const float* __restrict__ semB,
                              const float* __restrict__ semQ,
                              float* __restrict__ score_p) {
    const int m0   = blockIdx.x * 16;
    const int wave = threadIdx.x >> 5;
    const int lane = threadIdx.x & 31;
    const int c0   = wave * 16;
    const int half = lane >> 4;
    const int l16  = lane & 15;
    const int arow = m0 + l16;
    const int bcol = c0 + l16;

    v8f c = {};
    #pragma unroll
    for (int k0 = 0; k0 < HF; k0 += 4) {
        const int ka = k0 + half * 2;
        v2f a, b;
        a.x = embeds_p[(size_t)arow * HF + ka];
        a.y = embeds_p[(size_t)arow * HF + ka + 1];
        b.x = semW[(size_t)ka * SEM_DIM + bcol];
        b.y = semW[(size_t)(ka + 1) * SEM_DIM + bcol];
        c = __builtin_amdgcn_wmma_f32_16x16x4_f32(
                false, a, false, b, (short)0, c, false, false);
    }
    const float sb = semB[bcol];
    const float sq = semQ[bcol];
    float acc = 0.f;
    #pragma unroll
    for (int j = 0; j < 8; ++j) acc += tanhf(c[j] + sb) * sq;
    #pragma unroll
    for (int m = 16; m >= 1; m >>= 1) acc += __shfl_xor(acc, m, 32);
    if (lane == 0) atomicAdd(score_p, acc);
}

// ---------------------------------------------------------------------------
// Kernel 9: beta = softmax(scores / N)
// ---------------------------------------------------------------------------
__global__ void beta_kernel(const float* __restrict__ scores, float* __restrict__ beta) {
    float s[N_PATHS];
    float m = -1e30f;
    for (int p = 0; p < N_PATHS; ++p) { s[p] = scores[p] * (1.0f / N_NODES); m = fmaxf(m, s[p]); }
    float sum = 0.f;
    for (int p = 0; p < N_PATHS; ++p) { s[p] = __expf(s[p] - m); sum += s[p]; }
    for (int p = 0; p < N_PATHS; ++p) beta[p] = s[p] / sum;
}

// ---------------------------------------------------------------------------
// Kernel 10: final = sum_p beta_p * tanh(embeds_p@semW + b); logits = final@linW
//            + linB; out = log_softmax(logits). Weights cached in LDS.
// ---------------------------------------------------------------------------
__global__ void final_kernel(const float* __restrict__ embeds,  // [P][N][64]
                             const float* __restrict__ semW,
                             const float* __restrict__ semB,
                             const float* __restrict__ linW,
                             const float* __restrict__ linB,
                             const float* __restrict__ beta,
                             float* __restrict__ out) {
    __shared__ float sW[HF * SEM_DIM];         // 32 KB
    __shared__ float sLW[SEM_DIM * N_CLASSES]; // 8 KB
    __shared__ float sB[SEM_DIM];
    __shared__ float sLB[N_CLASSES];
    __shared__ float sBeta[N_PATHS];
    for (int i = threadIdx.x; i < HF * SEM_DIM; i += blockDim.x) sW[i] = semW[i];
    for (int i = threadIdx.x; i < SEM_DIM * N_CLASSES; i += blockDim.x) sLW[i] = linW[i];
    for (int i = threadIdx.x; i < SEM_DIM; i += blockDim.x) sB[i] = semB[i];
    if (threadIdx.x < N_CLASSES) sLB[threadIdx.x] = linB[threadIdx.x];
    if (threadIdx.x < N_PATHS)   sBeta[threadIdx.x] = beta[threadIdx.x];
    __syncthreads();

    const int n = blockIdx.x * blockDim.x + threadIdx.x;
    if (n >= N_NODES) return;

    float logits[N_CLASSES];
    #pragma unroll
    for (int cc = 0; cc < N_CLASSES; ++cc) logits[cc] = sLB[cc];

    for (int p = 0; p < N_PATHS; ++p) {
        float emb[HF];
        const float* ep = embeds + ((size_t)p * N_NODES + n) * HF;
        #pragma unroll
        for (int f = 0; f < HF; ++f) emb[f] = ep[f];
        const float bp = sBeta[p];
        for (int j = 0; j < SEM_DIM; ++j) {
            float s = sB[j];
            #pragma unroll
            for (int f = 0; f < HF; ++f) s += emb[f] * sW[f * SEM_DIM + j];
            const float w = bp * tanhf(s);
            #pragma unroll
            for (int cc = 0; cc < N_CLASSES; ++cc) logits[cc] += w * sLW[j * N_CLASSES + cc];
        }
    }
    float m = logits[0];
    #pragma unroll
    for (int cc = 1; cc < N_CLASSES; ++cc) m = fmaxf(m, logits[cc]);
    float sum = 0.f;
    #pragma unroll
    for (int cc = 0; cc < N_CLASSES; ++cc) sum += __expf(logits[cc] - m);
    const float ls = __logf(sum);
    #pragma unroll
    for (int cc = 0; cc < N_CLASSES; ++cc)
        out[(size_t)n * N_CLASSES + cc] = logits[cc] - m - ls;
}

// ---------------------------------------------------------------------------
extern "C" void kernel_launch(void* const* d_in, const int* in_sizes, int n_in,
                              void* d_out, int out_size, void* d_ws, size_t ws_size,
                              hipStream_t stream) {
    const float* x      = (const float*)d_in[0];
    const int*   edges  = (const int*)d_in[1];   // [P][2][E]
    const float* gatW   = (const float*)d_in[2];
    const float* aSrc   = (const float*)d_in[3];
    const float* aDst   = (const float*)d_in[4];
    const float* gatB   = (const float*)d_in[5];
    const float* semW   = (const float*)d_in[6];
    const float* semB   = (const float*)d_in[7];
    const float* semQ   = (const float*)d_in[8];
    const float* linW   = (const float*)d_in[9];
    const float* linB   = (const float*)d_in[10];
    float* out = (float*)d_out;

    // workspace layout (bytes, all 256-aligned)
    char* ws = (char*)d_ws;
    float* h      = (float*)(ws);                 // 25,600,000
    float* a_s    = (float*)(ws + 25600000);      //  3,200,000
    float* a_d    = (float*)(ws + 28800000);      //  3,200,000
    int*   maxv   = (int*)  (ws + 32000000);      //  9,600,000 (3 paths)
    float* sumv   = (float*)(ws + 41600000);      //  9,600,000
    float* embeds = (float*)(ws + 51200000);      // 76,800,000
    float* scores = (float*)(ws + 128000000);     // 3 floats
    float* beta   = (float*)(ws + 128000256);     // 3 floats
    // optional streaming weight buffer (one path at a time): 35,200,000 bytes
    const size_t WBUF_OFF  = 128000512;
    const size_t WBUF_SIZE = (size_t)MTOT * HEADS * sizeof(float);
    float* wbuf = (ws_size >= WBUF_OFF + WBUF_SIZE) ? (float*)(ws + WBUF_OFF) : nullptr;

    // 1) h = x @ W (WMMA)
    gemm_h_kernel<<<N_NODES / 16, 128, 0, stream>>>(x, gatW, h);

    // 2) attention coefficients
    att_coeff_kernel<<<(N_NODES + 255) / 256, 256, 0, stream>>>(h, aSrc, aDst, a_s, a_d);

    // 3) init accumulators
    {
        const size_t tot = (size_t)N_PATHS * N_NODES * HF;
        init_kernel<<<(unsigned)((tot + 255) / 256), 256, 0, stream>>>(maxv, sumv, embeds, scores);
    }

    // 4-7) per-metapath GAT edge softmax + aggregation
    for (int p = 0; p < N_PATHS; ++p) {
        const int* src = edges + (size_t)p * 2 * N_EDGES;
        const int* dst = src + N_EDGES;
        int*   mv = maxv   + (size_t)p * N_NODES * HEADS;
        float* sv = sumv   + (size_t)p * N_NODES * HEADS;
        float* em = embeds + (size_t)p * N_NODES * HF;

        const unsigned gEH = (unsigned)(((size_t)MTOT * HEADS + 255) / 256);
        edge_max_kernel<<<gEH, 256, 0, stream>>>(src, dst, a_s, a_d, mv);
        edge_sum_kernel<<<gEH, 256, 0, stream>>>(src, dst, a_s, a_d, mv, sv, wbuf);

        if (wbuf) {
            const unsigned gEQ = (unsigned)(((size_t)MTOT * (HF / 4) + 255) / 256);
            edge_acc_w_kernel<<<gEQ, 256, 0, stream>>>(src, dst, wbuf, h, em);
        } else {
            const unsigned gEF = (unsigned)(((size_t)MTOT * HF + 255) / 256);
            edge_acc_kernel<<<gEF, 256, 0, stream>>>(src, dst, a_s, a_d, mv, h, em);
        }

        const unsigned gNF = (unsigned)(((size_t)N_NODES * HF + 255) / 256);
        edge_fin_kernel<<<gNF, 256, 0, stream>>>(sv, gatB, em);
    }

    // 8) semantic attention scores (WMMA)
    for (int p = 0; p < N_PATHS; ++p) {
        const float* em = embeds + (size_t)p * N_NODES * HF;
        scores_kernel<<<N_NODES / 16, 256, 0, stream>>>(em, semW, semB, semQ, scores + p);
    }

    // 9) beta softmax
    beta_kernel<<<1, 1, 0, stream>>>(scores, beta);

    // 10) final projection + log_softmax
    final_kernel<<<(N_NODES + 255) / 256, 256, 0, stream>>>(
        embeds, semW, semB, linW, linB, beta, out);
}